// G_Nbody_43379169689774
// MI455X (gfx1250) — compile-verified
//
#include <hip/hip_runtime.h>

typedef __attribute__((ext_vector_type(2))) float v2f;
typedef __attribute__((ext_vector_type(8))) float v8f;

#define G_CONSTF 0.01f
#define EPSF     1e-6f
#define NBODY    4096
#define JPAGE    1024            // j rows cached in LDS per page
#define BLOCK    128             // 4 waves of 32
#define WAVES    (BLOCK / 32)
#define ROWS_PER_BLOCK (WAVES * 16)

__global__ __launch_bounds__(BLOCK)
void nbody_wmma_kernel(const float* __restrict__ x,   // [N,6] = [q|p]
                       const float* __restrict__ m,   // [N,1]
                       float* __restrict__ out)       // [N,6]
{
    // component table: rows 0..2 = qx,qy,qz ; row 3 = 1.0 ; row 4 = 0.0
    __shared__ float qcomp[5][JPAGE];
    __shared__ float mj[JPAGE], n2[JPAGE];
    __shared__ float wl[WAVES][16 * 17];              // per-wave W / T tile (padded)

    const int tid  = threadIdx.x;
    const int lane = tid & 31;
    const int wave = tid >> 5;
    const bool lo  = (lane < 16);
    const int  nl  = lane & 15;
    const int  hi8 = lo ? 0 : 8;

    const int i0 = blockIdx.x * ROWS_PER_BLOCK + wave * 16;

    // ---- A operand for gram GEMM: Q_i tile (16x4 f32, col 3 = 0 pad) ----
    v2f a1;
    {
        const float* xr = x + (size_t)(i0 + nl) * 6;
        a1.x = lo ? xr[0] : xr[2];
        a1.y = lo ? xr[1] : 0.0f;
    }

    // ---- squared norms of my 8 i-rows (C/D layout: VGPR r -> row r + hi8) ----
    float ni2[8];
#pragma unroll
    for (int r = 0; r < 8; ++r) {
        const float* xr = x + (size_t)(i0 + r + hi8) * 6;
        ni2[r] = xr[0] * xr[0] + xr[1] * xr[1] + xr[2] * xr[2];
    }

    // per-lane component row for GEMM2 B operand: col nl -> qx/qy/qz/1/0
    const float* comp = &qcomp[(nl < 4) ? nl : 4][0];

    float* wls = &wl[wave][0];
    v8f acc = {};                        // cols 0..2 = T, col 3 = S

    for (int page = 0; page < NBODY; page += JPAGE) {
        // cooperative page load (uniform trip count: all lanes active)
        for (int idx = tid; idx < JPAGE; idx += BLOCK) {
            const float* xr = x + (size_t)(page + idx) * 6;
            float a = xr[0], b = xr[1], c = xr[2];
            qcomp[0][idx] = a; qcomp[1][idx] = b; qcomp[2][idx] = c;
            qcomp[3][idx] = 1.0f; qcomp[4][idx] = 0.0f;
            n2[idx] = a * a + b * b + c * c;
            mj[idx] = m[page + idx];
        }
        __syncthreads();

        if (page + JPAGE < NBODY)        // prefetch next page (global_prefetch_b8)
            __builtin_prefetch(x + (size_t)(page + JPAGE) * 6 + tid, 0, 0);

        for (int jc = 0; jc < JPAGE; jc += 16) {
            const int j0 = page + jc;

            // B operand for gram GEMM: Q_j^T (4x16)
            v2f b1;
            b1.x = lo ? qcomp[0][jc + nl] : qcomp[2][jc + nl];
            b1.y = lo ? qcomp[1][jc + nl] : 0.0f;

            v8f czero = {};
            v8f gram = __builtin_amdgcn_wmma_f32_16x16x4_f32(
                false, a1, false, b1, (short)0, czero, false, false);

            // elementwise: W tile -> per-wave LDS (D-layout -> row-major)
            const float n2j  = n2[jc + nl];
            const float mjj  = mj[jc + nl];
            const int   colj = j0 + nl;
#pragma unroll
            for (int r = 0; r < 8; ++r) {
                float sq = ni2[r] + n2j - 2.0f * gram[r];
                sq = sq > 0.0f ? sq : 0.0f;
                float dist = __builtin_amdgcn_sqrtf(sq) + EPSF;   // raw v_sqrt_f32
                float d3 = dist * dist * dist;
                float w = mjj * __builtin_amdgcn_rcpf(d3);        // raw v_rcp_f32
                if (i0 + r + hi8 == colj) w = 0.0f;               // kill diagonal
                wls[(r + hi8) * 17 + nl] = w;
            }
            asm volatile("s_wait_dscnt 0x0" ::: "memory");

            // GEMM2: acc += W (16x16) * [qx qy qz 1 0..0], split K=16 -> 4x K=4
#pragma unroll
            for (int k0 = 0; k0 < 16; k0 += 4) {
                v2f a2;
                a2.x = wls[nl * 17 + k0 + (lo ? 0 : 2)];
                a2.y = wls[nl * 17 + k0 + (lo ? 1 : 3)];
                const int kx = jc + k0 + (lo ? 0 : 2);
                const int ky = jc + k0 + (lo ? 1 : 3);
                v2f b2;
                b2.x = comp[kx];     // unconditional LDS load (table has 1/0 rows)
                b2.y = comp[ky];
                acc = __builtin_amdgcn_wmma_f32_16x16x4_f32(
                    false, a2, false, b2, (short)0, acc, false, false);
            }
        }
        __syncthreads();                 // protect page before overwrite
    }

    // ---- dump accumulator tile to LDS, then 16 lanes finalize 16 rows ----
#pragma unroll
    for (int r = 0; r < 8; ++r)
        wls[(r + hi8) * 17 + nl] = acc[r];
    asm volatile("s_wait_dscnt 0x0" ::: "memory");

    if (lo) {
        const int row = i0 + nl;
        const float Tx = wls[nl * 17 + 0];
        const float Ty = wls[nl * 17 + 1];
        const float Tz = wls[nl * 17 + 2];
        const float S  = wls[nl * 17 + 3];
        const float* xr = x + (size_t)row * 6;
        const float qxi = xr[0], qyi = xr[1], qzi = xr[2];
        const float px  = xr[3], py  = xr[4], pz  = xr[5];
        const float mi  = m[row];
        const float invm = 1.0f / mi;
        const float gm   = G_CONSTF * mi;
        float* o = out + (size_t)row * 6;
        o[0] = px * invm;
        o[1] = py * invm;
        o[2] = pz * invm;
        o[3] = gm * (Tx - qxi * S);      // -dotp = G*m_i*(T - q_i*S)
        o[4] = gm * (Ty - qyi * S);
        o[5] = gm * (Tz - qzi * S);
    }
}

extern "C" void kernel_launch(void* const* d_in, const int* in_sizes, int n_in,
                              void* d_out, int out_size, void* d_ws, size_t ws_size,
                              hipStream_t stream) {
    // setup_inputs order: d_in[0]=t (unused), d_in[1]=x [N,6], d_in[2]=m [N,1]
    const float* x = (const float*)d_in[1];
    const float* m = (const float*)d_in[2];
    float* out = (float*)d_out;

    dim3 grid(NBODY / ROWS_PER_BLOCK);   // 64 blocks
    dim3 block(BLOCK);                   // 128 threads = 4 waves
    nbody_wmma_kernel<<<grid, block, 0, stream>>>(x, m, out);
}